// HeteroGCN_71107478552876
// MI455X (gfx1250) — compile-verified
//
#include <hip/hip_runtime.h>

// ---------------------------------------------------------------------------
// HeteroGCN on MI455X (gfx1250): f32 WMMA (16x16x4) for dense transforms,
// L2-resident f32 atomics for segment-sum message passing.
// ---------------------------------------------------------------------------

typedef __attribute__((ext_vector_type(2))) float v2f;
typedef __attribute__((ext_vector_type(8))) float v8f;

static constexpr int NUc = 200000;
static constexpr int NIc = 100000;
static constexpr int Ec  = 1000000;
static constexpr int NQc = 100000;

// ---------------- utility kernels ----------------

__global__ void fill_f32(float* __restrict__ p, float v, long n4) {
  long t = (long)blockIdx.x * blockDim.x + threadIdx.x;
  if (t < n4) ((float4*)p)[t] = make_float4(v, v, v, v);
}

__global__ void degree_kernel(const int* __restrict__ src, const int* __restrict__ dst,
                              float* __restrict__ du, float* __restrict__ di, int ne) {
  int e = blockIdx.x * blockDim.x + threadIdx.x;
  if (e < ne) {
    atomicAdd(du + src[e], 1.0f);
    atomicAdd(di + dst[e], 1.0f);
  }
}

__global__ void rsqrt_kernel(float* __restrict__ p, long n) {
  long t = (long)blockIdx.x * blockDim.x + threadIdx.x;
  if (t < n) {
    float d = p[t];
    p[t] = (d > 0.0f) ? rsqrtf(d) : 0.0f;
  }
}

// norm[e] = inv_deg_u[src[e]] * inv_deg_i[dst[e]]  (same for both directions)
__global__ void norm_kernel(const int* __restrict__ src, const int* __restrict__ dst,
                            const float* __restrict__ iu, const float* __restrict__ ii,
                            float* __restrict__ nrm, int ne) {
  int e = blockIdx.x * blockDim.x + threadIdx.x;
  if (e < ne) nrm[e] = iu[src[e]] * ii[dst[e]];
}

// ---------------- dense transform: H[N,64] = X[N,64] @ W[64,64] -------------
// One 256-thread block = 8 waves; each wave computes a 16-row x 64-col tile
// with V_WMMA_F32_16X16X4_F32 (K accumulated over 16 steps of 4).
// W is staged in LDS (16 KB); B fragments come from ds_load.

__global__ void __launch_bounds__(256) transform64(const float* __restrict__ X,
                                                   const float* __restrict__ W,
                                                   float* __restrict__ H, int nrows) {
  __shared__ float Wl[64 * 64];
  const int tid = threadIdx.x;
  #pragma unroll
  for (int i = 0; i < 16; ++i) Wl[tid + i * 256] = W[tid + i * 256];
  __syncthreads();

  const int wave = tid >> 5, lane = tid & 31;
  const long row0 = (long)blockIdx.x * 128 + wave * 16;
  if (row0 >= nrows) return;   // nrows % 16 == 0, so stripes are all-or-nothing

  const int m = lane & 15;       // M row within tile
  const int kh = lane >> 4;      // K half (lanes 16-31 hold K+2 per ISA layout)

  // A fragments for all 16 K-blocks: a[kb] = X[row0+m][kb*4 + kh*2 + {0,1}]
  const float2* xr = (const float2*)(X + (row0 + m) * 64 + kh * 2);
  v2f a[16];
  #pragma unroll
  for (int kb = 0; kb < 16; ++kb) {
    float2 t = xr[kb * 2];
    a[kb][0] = t.x; a[kb][1] = t.y;
  }

  #pragma unroll
  for (int t = 0; t < 4; ++t) {          // 4 N-tiles of 16 columns
    const int n = t * 16 + m;
    v8f c = {};
    #pragma unroll
    for (int kb = 0; kb < 16; ++kb) {
      v2f b;
      b[0] = Wl[(kb * 4 + kh * 2) * 64 + n];
      b[1] = Wl[(kb * 4 + kh * 2 + 1) * 64 + n];
      c = __builtin_amdgcn_wmma_f32_16x16x4_f32(false, a[kb], false, b,
                                                (short)0, c, false, false);
    }
    // D layout: lane l holds column n, VGPR r holds row M = r + 8*kh
    float* hp = H + row0 * 64 + n;
    #pragma unroll
    for (int r = 0; r < 8; ++r) hp[(r + 8 * kh) * 64] = c[r];
  }
}

// ---------------- message passing scatter -----------------------------------
// thread -> (edge, 4 features). acc[scat[e]] += h[gath[e]] * norm[e]
// Accumulators are L2-resident (<= 51.2 MB), atomics resolve in L2.

__global__ void scatter_add(const int* __restrict__ gath, const int* __restrict__ scat,
                            const float* __restrict__ nrm, const float* __restrict__ h,
                            float* __restrict__ acc, int ne) {
  long t = (long)blockIdx.x * blockDim.x + threadIdx.x;
  long e = t >> 4;
  if (e >= ne) return;
  int jj = (int)(t & 15) * 4;
  int g = gath[e], s = scat[e];
  float nm = nrm[e];
  float4 v = *(const float4*)(h + (long)g * 64 + jj);
  float* ap = acc + (long)s * 64 + jj;
  atomicAdd(ap + 0, v.x * nm);
  atomicAdd(ap + 1, v.y * nm);
  atomicAdd(ap + 2, v.z * nm);
  atomicAdd(ap + 3, v.w * nm);
}

// out = max(out, acc + bias)   (layer1 'max' aggregation; ReLU folded via init=0)
__global__ void max_bias(float* __restrict__ out, const float* __restrict__ acc,
                         const float* __restrict__ b, long n4) {
  long t = (long)blockIdx.x * blockDim.x + threadIdx.x;
  if (t >= n4) return;
  int j = (int)(t & 15) * 4;
  float4 a = ((const float4*)acc)[t];
  float4 o = ((float4*)out)[t];
  o.x = fmaxf(o.x, a.x + b[j + 0]);
  o.y = fmaxf(o.y, a.y + b[j + 1]);
  o.z = fmaxf(o.z, a.z + b[j + 2]);
  o.w = fmaxf(o.w, a.w + b[j + 3]);
  ((float4*)out)[t] = o;
}

// out = acc + b[0]+b[1]+b[2] (row-wise)   (layer2 'sum' aggregation biases)
__global__ void add_bias3(float* __restrict__ out, const float* __restrict__ acc,
                          const float* __restrict__ b, long n4) {
  long t = (long)blockIdx.x * blockDim.x + threadIdx.x;
  if (t >= n4) return;
  int j = (int)(t & 15) * 4;
  float4 a = ((const float4*)acc)[t];
  float4 o;
  o.x = a.x + b[j + 0] + b[64 + j + 0] + b[128 + j + 0];
  o.y = a.y + b[j + 1] + b[64 + j + 1] + b[128 + j + 1];
  o.z = a.z + b[j + 2] + b[64 + j + 2] + b[128 + j + 2];
  o.w = a.w + b[j + 3] + b[64 + j + 3] + b[128 + j + 3];
  ((float4*)out)[t] = o;
}

// ---------------- decoder ----------------------------------------------------
// 128 threads = 4 waves; each wave handles 16 queries.
// Stage 1 (WMMA): h = relu([xu[uid] || xi[iid]] @ dec_w1 + dec_b1), K=128.
// Stage 2 (scalar via LDS): out = h @ dec_w2 + dec_b2 (64 -> 4).

__global__ void __launch_bounds__(128) decoder_kernel(
    const float* __restrict__ xu, const float* __restrict__ xi,
    const int* __restrict__ uids, const int* __restrict__ iids,
    const float* __restrict__ w1, const float* __restrict__ b1,
    const float* __restrict__ w2, const float* __restrict__ b2,
    float* __restrict__ out, int nq) {
  __shared__ float W1l[128 * 64];        // 32 KB
  __shared__ float Hl[4][16 * 65];       // padded stride to dodge bank conflicts
  const int tid = threadIdx.x;
  #pragma unroll
  for (int i = 0; i < 64; ++i) W1l[tid + i * 128] = w1[tid + i * 128];
  __syncthreads();

  const int wave = tid >> 5, lane = tid & 31;
  const int q0 = blockIdx.x * 64 + wave * 16;
  if (q0 >= nq) return;                  // NQ % 16 == 0

  const int m = lane & 15, kh = lane >> 4;
  const int uid = uids[q0 + m], iid = iids[q0 + m];

  // A fragments: 32 K-blocks (K=128): first 64 from user row, next 64 from item row
  v2f a[32];
  const float2* ur = (const float2*)(xu + (long)uid * 64 + kh * 2);
  const float2* ir = (const float2*)(xi + (long)iid * 64 + kh * 2);
  #pragma unroll
  for (int kb = 0; kb < 16; ++kb) {
    float2 t = ur[kb * 2]; a[kb][0] = t.x; a[kb][1] = t.y;
  }
  #pragma unroll
  for (int kb = 0; kb < 16; ++kb) {
    float2 t = ir[kb * 2]; a[16 + kb][0] = t.x; a[16 + kb][1] = t.y;
  }

  float* hw = &Hl[wave][0];
  #pragma unroll
  for (int t = 0; t < 4; ++t) {
    const int n = t * 16 + m;
    v8f c = {};
    #pragma unroll
    for (int kb = 0; kb < 32; ++kb) {
      v2f b;
      b[0] = W1l[(kb * 4 + kh * 2) * 64 + n];
      b[1] = W1l[(kb * 4 + kh * 2 + 1) * 64 + n];
      c = __builtin_amdgcn_wmma_f32_16x16x4_f32(false, a[kb], false, b,
                                                (short)0, c, false, false);
    }
    const float bias = b1[n];
    #pragma unroll
    for (int r = 0; r < 8; ++r) {
      float v = c[r] + bias;
      hw[(r + 8 * kh) * 65 + n] = (v > 0.0f) ? v : 0.0f;  // same-wave LDS, in-order
    }
  }

  // Stage 2: each lane produces 2 of the 64 (query, out-channel) results
  #pragma unroll
  for (int half = 0; half < 2; ++half) {
    int idx = lane + 32 * half;          // 0..63
    int qq = idx >> 2, o = idx & 3;
    float s = b2[o];
    #pragma unroll 8
    for (int j = 0; j < 64; ++j) s += hw[qq * 65 + j] * w2[j * 4 + o];
    out[(long)(q0 + qq) * 4 + o] = s;
  }
}

// ---------------------------------------------------------------------------

extern "C" void kernel_launch(void* const* d_in, const int* in_sizes, int n_in,
                              void* d_out, int out_size, void* d_ws, size_t ws_size,
                              hipStream_t stream) {
  (void)in_sizes; (void)n_in; (void)out_size; (void)ws_size;
  const float* user_emb = (const float*)d_in[0];
  const float* item_emb = (const float*)d_in[1];
  const float* W1 = (const float*)d_in[2];
  const float* b1 = (const float*)d_in[3];
  const float* W2 = (const float*)d_in[4];
  const float* b2 = (const float*)d_in[5];
  const float* dw1 = (const float*)d_in[6];
  const float* db1 = (const float*)d_in[7];
  const float* dw2 = (const float*)d_in[8];
  const float* db2 = (const float*)d_in[9];
  const int* esrc[3] = { (const int*)d_in[10], (const int*)d_in[11], (const int*)d_in[12] };
  const int* edst[3] = { esrc[0] + Ec, esrc[1] + Ec, esrc[2] + Ec };
  const int* uids = (const int*)d_in[13];
  const int* iids = (const int*)d_in[14];
  float* out = (float*)d_out;

  // workspace carve-up (f32): ~272 MB total
  float* ws = (float*)d_ws;
  float* inv_u = ws;                       // 3 * NU
  float* inv_i = inv_u + 3L * NUc;         // 3 * NI
  float* nrm   = inv_i + 3L * NIc;         // 3 * E
  float* h_buf = nrm   + 3L * Ec;          // NU * 64
  float* acc   = h_buf + (long)NUc * 64;   // NU * 64
  float* xu_a  = acc   + (long)NUc * 64;   // NU * 64 (layer1 user out)
  float* xi_a  = xu_a  + (long)NUc * 64;   // NI * 64 (layer1 item out)
  float* xu_b  = xi_a  + (long)NIc * 64;   // NU * 64 (layer2 user out)
  float* xi_b  = xu_b  + (long)NUc * 64;   // NI * 64 (layer2 item out)

  auto fills = [&](float* p, float v, long n) {
    long n4 = n / 4;
    fill_f32<<<(int)((n4 + 255) / 256), 256, 0, stream>>>(p, v, n4);
  };
  const int egrid = (Ec + 255) / 256;
  const int sgrid = (int)((16L * Ec) / 256);      // 62500
  auto xform = [&](const float* X, const float* W, int n) {
    transform64<<<(n + 127) / 128, 256, 0, stream>>>(X, W, h_buf, n);
  };

  // ---- Phase 0: degrees -> inv_sqrt -> per-edge norms (shared by both layers)
  fills(inv_u, 0.0f, 3L * NUc);
  fills(inv_i, 0.0f, 3L * NIc);
  for (int k = 0; k < 3; ++k)
    degree_kernel<<<egrid, 256, 0, stream>>>(esrc[k], edst[k],
                                             inv_u + (long)k * NUc,
                                             inv_i + (long)k * NIc, Ec);
  rsqrt_kernel<<<(int)((3L * NUc + 255) / 256), 256, 0, stream>>>(inv_u, 3L * NUc);
  rsqrt_kernel<<<(int)((3L * NIc + 255) / 256), 256, 0, stream>>>(inv_i, 3L * NIc);
  for (int k = 0; k < 3; ++k)
    norm_kernel<<<egrid, 256, 0, stream>>>(esrc[k], edst[k],
                                           inv_u + (long)k * NUc,
                                           inv_i + (long)k * NIc,
                                           nrm + (long)k * Ec, Ec);

  // ---- Layer 1, item side: relations 0..2 (user->item), max aggr (+ReLU folded)
  fills(xi_a, 0.0f, (long)NIc * 64);
  for (int k = 0; k < 3; ++k) {
    xform(user_emb, W1 + (long)k * 4096, NUc);
    fills(acc, 0.0f, (long)NIc * 64);
    scatter_add<<<sgrid, 256, 0, stream>>>(esrc[k], edst[k], nrm + (long)k * Ec,
                                           h_buf, acc, Ec);
    max_bias<<<(int)(((long)NIc * 16 + 255) / 256), 256, 0, stream>>>(
        xi_a, acc, b1 + (long)k * 64, (long)NIc * 16);
  }
  // ---- Layer 1, user side: relations 3..5 (item->user)
  fills(xu_a, 0.0f, (long)NUc * 64);
  for (int k = 0; k < 3; ++k) {
    xform(item_emb, W1 + (long)(k + 3) * 4096, NIc);
    fills(acc, 0.0f, (long)NUc * 64);
    scatter_add<<<sgrid, 256, 0, stream>>>(edst[k], esrc[k], nrm + (long)k * Ec,
                                           h_buf, acc, Ec);
    max_bias<<<(int)(((long)NUc * 16 + 255) / 256), 256, 0, stream>>>(
        xu_a, acc, b1 + (long)(k + 3) * 64, (long)NUc * 16);
  }

  // ---- Layer 2, item side: sum aggr -> accumulate all relations into one buffer
  fills(acc, 0.0f, (long)NIc * 64);
  for (int k = 0; k < 3; ++k) {
    xform(xu_a, W2 + (long)k * 4096, NUc);
    scatter_add<<<sgrid, 256, 0, stream>>>(esrc[k], edst[k], nrm + (long)k * Ec,
                                           h_buf, acc, Ec);
  }
  add_bias3<<<(int)(((long)NIc * 16 + 255) / 256), 256, 0, stream>>>(
      xi_b, acc, b2, (long)NIc * 16);
  // ---- Layer 2, user side
  fills(acc, 0.0f, (long)NUc * 64);
  for (int k = 0; k < 3; ++k) {
    xform(xi_a, W2 + (long)(k + 3) * 4096, NIc);
    scatter_add<<<sgrid, 256, 0, stream>>>(edst[k], esrc[k], nrm + (long)k * Ec,
                                           h_buf, acc, Ec);
  }
  add_bias3<<<(int)(((long)NUc * 16 + 255) / 256), 256, 0, stream>>>(
      xu_b, acc, b2 + 3L * 64, (long)NUc * 16);

  // ---- Decoder
  decoder_kernel<<<(NQc + 63) / 64, 128, 0, stream>>>(
      xu_b, xi_b, uids, iids, dw1, db1, dw2, db2, out, NQc);
}